// TelluricModel_86225763435286
// MI455X (gfx1250) — compile-verified
//
#include <hip/hip_runtime.h>
#include <hip/hip_bf16.h>

// Sizes fixed by the reference harness.
#define BATCH    32
#define N_ORDER  64
#define N_SPEC   4096
#define N_LATENT 6

typedef float v2f __attribute__((ext_vector_type(2)));
typedef float v4f __attribute__((ext_vector_type(4)));
typedef float v8f __attribute__((ext_vector_type(8)));

// ---------------------------------------------------------------------------
// Kernel A: x_lines[b, r] = sin( sum_l s[b,l] * W[r,l] + bias[r] )
// D = A x B with V_WMMA_F32_16X16X4_F32:
//   A = s  [32 x 6]  (two 16-row M-tiles, K padded 6 -> 8, two K-steps of 4)
//   B = W^T [6 x N_REST] (each wave owns a 16-column chunk)
// f32 A-matrix 16x4 layout: lanes 0-15 hold K={0,1}, lanes 16-31 hold K={2,3}
// (2 VGPRs).  B 4x16 mirrors this with N striped across lanes.
// C/D 16x16 f32: VGPR r -> row r (lanes 0-15) / row r+8 (lanes 16-31).
//
// Every operand pair sits at an even float offset (row stride 6, pair base
// 0/2/4), i.e. 8-byte aligned -> load as v2f (global_load_b64), then zero the
// K=6,7 pad lanes with a vector select (v_cndmask, no EXEC branches near the
// WMMAs).  sin() arguments are |x| < ~0.03 (W,b ~ N(0,1e-3), K=6), so the
// hardware v_sin_f32 via __sinf is fp32-exact here.
// ---------------------------------------------------------------------------
__global__ __launch_bounds__(256) void decode_gemm_wmma(
    const float* __restrict__ s,      // [32, 6]
    const float* __restrict__ W,      // [n_rest, 6] row-major
    const float* __restrict__ bias,   // [n_rest]
    float*       __restrict__ xlines, // [32, n_rest]
    int n_rest)
{
    const int lane = threadIdx.x & 31;
    const int wave = threadIdx.x >> 5;
    const int n0   = (blockIdx.x * 8 + wave) * 16;   // 16-column chunk per wave
    if (n0 >= n_rest) return;                         // wave-uniform: EXEC stays full

    const int  half = lane >> 4;        // 0: lanes 0-15, 1: lanes 16-31
    const int  l15  = lane & 15;
    const bool lo   = (half == 0);      // lanes holding K={0,1}/{4,5}
    const v2f  zero = {0.0f, 0.0f};

    // ---- A operands from s: row = mtile*16 + l15; chunk0 K=0..3, chunk1 K=4..7 (6,7 zero)
    const float* p0 = s + l15 * N_LATENT;         // row of M-tile 0
    const float* p1 = s + (16 + l15) * N_LATENT;  // row of M-tile 1
    const v2f a0c0 = *(const v2f*)(p0 + half * 2);
    const v2f a1c0 = *(const v2f*)(p1 + half * 2);
    const v2f a0c1 = lo ? *(const v2f*)(p0 + 4) : zero;
    const v2f a1c1 = lo ? *(const v2f*)(p1 + 4) : zero;

    // ---- B operands: column n = n0 + l15; B[k][n] = W[n*6 + k]
    const int    n  = n0 + l15;
    const float* pw = W + (size_t)n * N_LATENT;
    const v2f b_c0 = *(const v2f*)(pw + half * 2);
    const v2f b_c1 = lo ? *(const v2f*)(pw + 4) : zero;

    v8f acc0 = {};
    v8f acc1 = {};
    // (neg_a, A, neg_b, B, c_mod, C, reuse_a, reuse_b)
    acc0 = __builtin_amdgcn_wmma_f32_16x16x4_f32(false, a0c0, false, b_c0, (short)0, acc0, false, false);
    acc0 = __builtin_amdgcn_wmma_f32_16x16x4_f32(false, a0c1, false, b_c1, (short)0, acc0, false, false);
    acc1 = __builtin_amdgcn_wmma_f32_16x16x4_f32(false, a1c0, false, b_c0, (short)0, acc1, false, false);
    acc1 = __builtin_amdgcn_wmma_f32_16x16x4_f32(false, a1c1, false, b_c1, (short)0, acc1, false, false);

    const float bb = bias[n];
    #pragma unroll
    for (int r = 0; r < 8; ++r) {
        const int row = r + half * 8;  // row within the 16-row M-tile
        xlines[(size_t)row        * n_rest + n] = __sinf(acc0[r] + bb);
        xlines[(size_t)(row + 16) * n_rest + n] = __sinf(acc1[r] + bb);
    }
}

// ---------------------------------------------------------------------------
// Kernel B: Catmull-Rom interpolation of x_lines at redshifted wavelengths.
// wave_rest is a uniform linspace, so searchsorted collapses to
// k = floor((xq - x0)/dx) with a +/-1 fix-up against the actual knot values
// (reproduces searchsorted(right)-1 exactly), clamped to [1, N-3].
// The 4 knots and 4 spectrum samples are contiguous 4B-aligned spans ->
// single 128-bit gathers; the fix-up reuses the loaded knot vector, so the
// common case is exactly 2 wide loads per output.  All gathers hit L2
// (wave_rest + x_lines ~26 MB << 192 MB).
// ---------------------------------------------------------------------------
__global__ __launch_bounds__(256) void interp_cubic(
    const float* __restrict__ z,         // [B, N_ORDER]
    const float* __restrict__ wave_rest, // [n_rest]
    const float* __restrict__ wave_obs,  // [N_ORDER, N_SPEC]
    const float* __restrict__ xlines,    // [B, n_rest]
    float*       __restrict__ out,       // [B, N_ORDER, N_SPEC]
    int n_rest)
{
    const long long idx   = (long long)blockIdx.x * blockDim.x + threadIdx.x;
    const long long total = (long long)BATCH * N_ORDER * N_SPEC;
    if (idx >= total) return;

    const int sidx = (int)(idx & (N_SPEC - 1));
    const int o    = (int)((idx >> 12) & (N_ORDER - 1));
    const int b    = (int)(idx >> 18);

    const float wobs = wave_obs[o * N_SPEC + sidx];
    const float zz   = z[b * N_ORDER + o];
    const float xq   = wobs * (1.0f - zz);

    // Uniform-grid index guess (uniform loads -> SMEM).
    const float xf     = wave_rest[0];
    const float inv_dx = 1.0f / (wave_rest[1] - xf);
    int k = (int)floorf((xq - xf) * inv_dx);
    if (k < 1) k = 1;
    if (k > n_rest - 3) k = n_rest - 3;

    // One b128 load of knots [k-1 .. k+2]; fix up k against the loaded values.
    // xx[1] = wave_rest[k], xx[2] = wave_rest[k+1].
    v4f xx;
    for (;;) {
        __builtin_memcpy(&xx, wave_rest + (k - 1), sizeof(v4f));
        if (k < n_rest - 3 && xx[2] <= xq) { ++k; continue; }
        if (k > 1 && xx[1] > xq)           { --k; continue; }
        break;
    }

    v4f yy;  // [ y(k-1), y(k), y(k+1), y(k+2) ]
    __builtin_memcpy(&yy, xlines + (size_t)b * n_rest + (k - 1), sizeof(v4f));

    const float h  = xx[2] - xx[1];
    const float t  = (xq - xx[1]) / h;
    const float t2 = t * t;
    const float t3 = t2 * t;
    const float h00 =  2.0f * t3 - 3.0f * t2 + 1.0f;
    const float h10 =         t3 - 2.0f * t2 + t;
    const float h01 = -2.0f * t3 + 3.0f * t2;
    const float h11 =         t3 -        t2;
    const float m0  = (yy[2] - yy[0]) / (xx[2] - xx[0]) * h;
    const float m1  = (yy[3] - yy[1]) / (xx[3] - xx[1]) * h;

    const float val = h00 * yy[1] + h10 * m0 + h01 * yy[2] + h11 * m1;
    out[idx] = 1.0f - val;
}

extern "C" void kernel_launch(void* const* d_in, const int* in_sizes, int n_in,
                              void* d_out, int out_size, void* d_ws, size_t ws_size,
                              hipStream_t stream) {
    const float* s         = (const float*)d_in[0]; // [32, 6]
    const float* z         = (const float*)d_in[1]; // [32, 64]
    const float* W         = (const float*)d_in[2]; // [n_rest, 6]
    const float* bias      = (const float*)d_in[3]; // [n_rest]
    const float* wave_rest = (const float*)d_in[4]; // [n_rest]
    const float* wave_obs  = (const float*)d_in[5]; // [64, 4096]
    float* out    = (float*)d_out;
    float* xlines = (float*)d_ws;                   // BATCH * n_rest floats (25.6 MB)

    const int n_rest  = in_sizes[4];
    const int nchunks = (n_rest + 15) / 16;          // 16-column chunks, one per wave
    const int blocksA = (nchunks + 7) / 8;           // 8 waves per 256-thread block
    decode_gemm_wmma<<<blocksA, 256, 0, stream>>>(s, W, bias, xlines, n_rest);

    const long long total = (long long)BATCH * N_ORDER * N_SPEC;
    const int blocksB = (int)((total + 255) / 256);
    interp_cubic<<<blocksB, 256, 0, stream>>>(z, wave_rest, wave_obs, xlines, out, n_rest);
}